// ScatterLoss_69217692942358
// MI455X (gfx1250) — compile-verified
//
#include <hip/hip_runtime.h>
#include <math.h>

// Problem constants (from the reference)
#define NSAMP  65536
#define DDIM   512
#define NCLS   512
#define MARGIN 1.0f
#define EPSV   1e-6f

// Decomposition
#define W       32              // d-columns per slice
#define SLICES  (DDIM / W)      // 16
#define CHUNKS  16              // sample chunks
#define CHUNK1  (NSAMP / CHUNKS) // 4096 samples per chunk

typedef unsigned int u32;
typedef unsigned long long u64;
typedef u32 u32x4 __attribute__((ext_vector_type(4)));
typedef int i32x4 __attribute__((ext_vector_type(4)));
typedef int i32x8 __attribute__((ext_vector_type(8)));
typedef float v2f __attribute__((ext_vector_type(2)));
typedef float v8f __attribute__((ext_vector_type(8)));

// ---------------------------------------------------------------------------
// counts
// ---------------------------------------------------------------------------
__global__ void k_zero(int* counts) {
    counts[threadIdx.x] = 0;
}

__global__ __launch_bounds__(256) void k_count(const int* __restrict__ labels,
                                               int* __restrict__ counts) {
    __shared__ int sh[NCLS];
    const int t = threadIdx.x;
    sh[t] = 0; sh[t + 256] = 0;
    __syncthreads();
    const int start = blockIdx.x * 1024;
    for (int i = start + t; i < start + 1024; i += 256)
        atomicAdd(&sh[labels[i] & (NCLS - 1)], 1);
    __syncthreads();
    atomicAdd(&counts[t],       sh[t]);
    atomicAdd(&counts[t + 256], sh[t + 256]);
}

// ---------------------------------------------------------------------------
// k1: privatized segment sum.  grid = SLICES*CHUNKS blocks, 256 threads.
// Block (s,h): accumulates d-columns [s*32, s*32+32) of samples in chunk h
// into a 512x32 LDS accumulator via ds_add_f32, writes partials to ws.
// Labels for the chunk are DMA'd into LDS by the Tensor Data Mover.
// ---------------------------------------------------------------------------
__global__ __launch_bounds__(256) void k1_partials(const float* __restrict__ outp,
                                                   const int* __restrict__ labels,
                                                   float* __restrict__ partials) {
    __shared__ float acc[NCLS * W];   // 64 KB
    __shared__ int   lbl[CHUNK1];     // 16 KB

    const int s = blockIdx.x / CHUNKS;
    const int h = blockIdx.x % CHUNKS;
    const int t = threadIdx.x;

    for (int i = t; i < NCLS * W; i += 256) acc[i] = 0.0f;

#if defined(__HIP_DEVICE_COMPILE__) && __has_builtin(__builtin_amdgcn_tensor_load_to_lds)
    if (t < 32) {
        // Build Tensor DMA descriptor (D#): contiguous copy of CHUNK1 dwords
        // (chunk's int32 labels) from global to LDS.
        const u64 ga = (u64)(const void*)(labels + (size_t)h * CHUNK1);
        const u32 lds_off = (u32)(u64)(void*)lbl; // flat addr low 32 bits == LDS byte offset
        u32x4 g0 = { 1u,                                   // count=1 (valid), user mode
                     lds_off,                               // lds_addr
                     (u32)ga,                               // global_addr[31:0]
                     ((u32)((ga >> 32) & 0x1FFFFFFu)) | (2u << 30) }; // [56:32] + type=2
        i32x8 g1 = { 0x00020000,   // wg_mask=0, data_size=2 (4B), no pad/iterate
                     0x10000000,   // tensor_dim0 = 4096  (bits[79:48], low 16 here)
                     0x00010000,   // tensor_dim0 hi=0 | tensor_dim1 = 1
                     0x10000000,   // tensor_dim1 hi=0 | tile_dim0 = 4096
                     0x00000001,   // tile_dim1 = 1, tile_dim2 = 0 (unused)
                     4096,         // tensor_dim0_stride = 4096
                     0, 0 };       // stride hi / tensor_dim1_stride = 0
        i32x4 g2 = { 1, 1, 0, 0 };           // tensor_dim2=1, tensor_dim3=1 (safety)
        i32x4 g3 = { 0, 0x00010000, 0, 0 };  // tensor_dim4=1 (safety)
        i32x8 g4 = { 0, 0, 0, 0, 0, 0, 0, 0 }; // unused/NULL extension group (VADDR4)
        __builtin_amdgcn_tensor_load_to_lds(g0, g1, g2, g3, g4, 0);
        __builtin_amdgcn_s_wait_tensorcnt(0);
    }
#else
    for (int i = t; i < CHUNK1; i += 256) lbl[i] = labels[(size_t)h * CHUNK1 + i];
#endif
    __syncthreads();

    const int wave = t >> 5;   // wave32
    const int lane = t & 31;
    const float* src = outp + ((size_t)h * CHUNK1) * DDIM + (size_t)s * W + lane;

    // 8 waves each walk their residue class; 4-way unroll for load ILP.
    for (int i = wave; i < CHUNK1; i += 32) {
        const int c0 = lbl[i]      & (NCLS - 1);
        const int c1 = lbl[i + 8]  & (NCLS - 1);
        const int c2 = lbl[i + 16] & (NCLS - 1);
        const int c3 = lbl[i + 24] & (NCLS - 1);
        const float v0 = src[(size_t)i * DDIM];
        const float v1 = src[(size_t)(i + 8)  * DDIM];
        const float v2 = src[(size_t)(i + 16) * DDIM];
        const float v3 = src[(size_t)(i + 24) * DDIM];
        atomicAdd(&acc[c0 * W + lane], v0);   // ds_add_f32, conflict-free banks
        atomicAdd(&acc[c1 * W + lane], v1);
        atomicAdd(&acc[c2 * W + lane], v2);
        atomicAdd(&acc[c3 * W + lane], v3);
    }
    __syncthreads();

    float* dst = partials + (size_t)blockIdx.x * (NCLS * W);
    for (int i = t; i < NCLS * W; i += 256) dst[i] = acc[i];
}

// ---------------------------------------------------------------------------
// k2: reduce 16 chunk-partials -> cls_sums, as a ones-vector contraction
// done with chained V_WMMA_F32_16X16X4_F32 (A = all-ones -> layout-invariant).
// grid = SLICES*8 blocks, 256 threads (8 waves). Each wave: 16 chains of
// 16 outputs x 16 chunks.
// ---------------------------------------------------------------------------
__global__ __launch_bounds__(256) void k2_reduce(const float* __restrict__ partials,
                                                 float* __restrict__ cls_sums) {
    const int s    = blockIdx.x >> 3;   // slice
    const int bIn  = blockIdx.x & 7;
    const int wave = threadIdx.x >> 5;
    const int lane = threadIdx.x & 31;
    const int waveInSlice = bIn * 8 + wave;   // 0..63

#if defined(__HIP_DEVICE_COMPILE__) && __has_builtin(__builtin_amdgcn_wmma_f32_16x16x4_f32)
    v2f a; a.x = 1.0f; a.y = 1.0f;            // all-ones A matrix
    const int n    = lane & 15;
    const int koff = (lane < 16) ? 0 : 2;
    for (int chain = waveInSlice; chain < 1024; chain += 64) {
        const int base = chain * 16;
        v8f c = {0.0f,0.0f,0.0f,0.0f,0.0f,0.0f,0.0f,0.0f};
        for (int q = 0; q < 4; ++q) {
            const int k0 = q * 4 + koff;
            v2f b;
            b.x = partials[((size_t)(s * CHUNKS + k0    )) * (NCLS * W) + base + n];
            b.y = partials[((size_t)(s * CHUNKS + k0 + 1)) * (NCLS * W) + base + n];
            c = __builtin_amdgcn_wmma_f32_16x16x4_f32(false, a, false, b,
                                                      (short)0, c, false, false);
        }
        if (lane < 16) {
            const int o  = base + lane;       // D row M=0: VGPR0, lanes 0..15, N=lane
            const int cc = o >> 5;
            const int dl = o & 31;
            cls_sums[cc * DDIM + s * W + dl] = c[0];
        }
    }
#else
    for (int o = bIn * 256 + (int)threadIdx.x; o < NCLS * W; o += 2048) {
        float v = 0.0f;
        for (int hh = 0; hh < CHUNKS; ++hh)
            v += partials[((size_t)(s * CHUNKS + hh)) * (NCLS * W) + o];
        const int cc = o >> 5, dl = o & 31;
        cls_sums[cc * DDIM + s * W + dl] = v;
    }
    (void)waveInSlice; (void)lane;
#endif
}

// ---------------------------------------------------------------------------
// total[d] = sum over classes of cls_sums[c][d].  grid = 16 x 256.
// ---------------------------------------------------------------------------
__global__ __launch_bounds__(256) void k_total(const float* __restrict__ cls_sums,
                                               float* __restrict__ total) {
    __shared__ float sh[256];
    const int t  = threadIdx.x;
    const int dl = t & 31;
    const int g  = t >> 5;                 // 8 class groups
    const int d  = blockIdx.x * 32 + dl;
    float acc = 0.0f;
    for (int c = g * 64; c < g * 64 + 64; ++c) acc += cls_sums[c * DDIM + d];
    sh[t] = acc;
    __syncthreads();
    if (t < 32) {
        float v = 0.0f;
        for (int gg = 0; gg < 8; ++gg) v += sh[gg * 32 + t];
        total[blockIdx.x * 32 + t] = v;
    }
}

// ---------------------------------------------------------------------------
// Per-class loss term: cnt_c * relu(1 - d_c)^2.  grid = 512 x 128.
// ---------------------------------------------------------------------------
__global__ __launch_bounds__(128) void k_loss(const float* __restrict__ cls_sums,
                                              const float* __restrict__ total,
                                              const int* __restrict__ counts,
                                              float* __restrict__ loss_c) {
    __shared__ float sh[128];
    const int c = blockIdx.x;
    const int t = threadIdx.x;
    const float cnt = (float)counts[c];
    float acc = 0.0f;
    if (cnt > 0.0f) {
        const float inv_c  = 1.0f / cnt;
        const float inv_nc = 1.0f / ((float)NSAMP - cnt);
        for (int d = t; d < DDIM; d += 128) {
            const float sm   = cls_sums[c * DDIM + d];
            const float diff = sm * inv_c - (total[d] - sm) * inv_nc + EPSV;
            acc += diff * diff;
        }
    }
    sh[t] = acc;
    __syncthreads();
    for (int off = 64; off > 0; off >>= 1) {
        if (t < off) sh[t] += sh[t + off];
        __syncthreads();
    }
    if (t == 0) {
        const float dd = sqrtf(sh[0]);
        const float m  = fmaxf(MARGIN - dd, 0.0f);
        loss_c[c] = (cnt > 0.0f) ? cnt * m * m : 0.0f;
    }
}

// ---------------------------------------------------------------------------
// Final mean. 1 x 256, fixed-order tree -> deterministic.
// ---------------------------------------------------------------------------
__global__ __launch_bounds__(256) void k_final(const float* __restrict__ loss_c,
                                               float* __restrict__ outp) {
    __shared__ float sh[256];
    const int t = threadIdx.x;
    sh[t] = loss_c[t] + loss_c[t + 256];
    __syncthreads();
    for (int off = 128; off > 0; off >>= 1) {
        if (t < off) sh[t] += sh[t + off];
        __syncthreads();
    }
    if (t == 0) outp[0] = sh[0] / (float)NSAMP;
}

// ---------------------------------------------------------------------------
extern "C" void kernel_launch(void* const* d_in, const int* in_sizes, int n_in,
                              void* d_out, int out_size, void* d_ws, size_t ws_size,
                              hipStream_t stream) {
    const float* outp   = (const float*)d_in[0];
    const int*   labels = (const int*)d_in[1];

    float* ws       = (float*)d_ws;
    float* partials = ws;                                               // 16 MB
    float* cls_sums = ws + (size_t)SLICES * CHUNKS * NCLS * W;          // +1 MB
    float* total    = cls_sums + (size_t)NCLS * DDIM;                   // +2 KB
    int*   counts   = (int*)(total + DDIM);                             // +2 KB
    float* loss_c   = (float*)(counts + NCLS);                          // +2 KB
    float* res      = (float*)d_out;

    k_zero<<<dim3(1), dim3(NCLS), 0, stream>>>(counts);
    k_count<<<dim3(NSAMP / 1024), dim3(256), 0, stream>>>(labels, counts);
    k1_partials<<<dim3(SLICES * CHUNKS), dim3(256), 0, stream>>>(outp, labels, partials);
    k2_reduce<<<dim3(SLICES * 8), dim3(256), 0, stream>>>(partials, cls_sums);
    k_total<<<dim3(SLICES), dim3(256), 0, stream>>>(cls_sums, total);
    k_loss<<<dim3(NCLS), dim3(128), 0, stream>>>(cls_sums, total, counts, loss_c);
    k_final<<<dim3(1), dim3(256), 0, stream>>>(loss_c, res);
    (void)in_sizes; (void)n_in; (void)out_size; (void)ws_size;
}